// CorrespodenceNet_29566554866195
// MI455X (gfx1250) — compile-verified
//
#include <hip/hip_runtime.h>
#include <math.h>

typedef __attribute__((ext_vector_type(2))) float v2f;
typedef __attribute__((ext_vector_type(8))) float v8f;

#define NPIX 8192
#define NCH  256
#define TAU_INV 100.0f
#define JT 64            // columns (j) per workgroup, resident in LDS
#define IT 32            // rows (i) per inner tile
#define CP (NCH + 4)     // padded row stride: +4 floats -> conflict-free b64 LDS loads

// ---------------------------------------------------------------------------
// float atomic max via signed/unsigned int trick (works with -inf init)
__device__ inline void atomicMaxFloat(float* addr, float val) {
    if (val >= 0.0f) atomicMax((int*)addr, __float_as_int(val));
    else             atomicMin((unsigned int*)addr, __float_as_uint(val));
}

// CDNA5 async global->LDS copy (ASYNCcnt path), 16B per lane
__device__ inline void async_ld_b128(unsigned lds_byte_addr, const void* gaddr) {
    asm volatile("global_load_async_to_lds_b128 %0, %1, off"
                 :: "v"(lds_byte_addr), "v"(gaddr) : "memory");
}
__device__ inline void wait_async0() {
    asm volatile("s_wait_asynccnt 0x0" ::: "memory");
}

// ---------------------------------------------------------------------------
__global__ void init_kernel(float* out, float* ws) {
    int i = blockIdx.x * 256 + threadIdx.x;
    if (i < 256)            out[i] = 0.0f;              // warped_color accumulators
    else if (i < 256+NPIX)  out[i] = -__builtin_inff(); // confidence (row-max) init
    if (i < 512)            ws[i]  = 0.0f;              // sum-of-squares accumulators
}

// column sum of squares (reduce over N rows); coalesced: thread c reads row[c]
__global__ void colsumsq_kernel(const float* __restrict__ X,
                                const float* __restrict__ Y,
                                float* __restrict__ sumx,
                                float* __restrict__ sumy) {
    const float* src = blockIdx.y ? Y : X;
    float*       dst = blockIdx.y ? sumy : sumx;
    int c  = threadIdx.x;
    int r0 = blockIdx.x * 128;
    float acc = 0.0f;
    for (int r = 0; r < 128; ++r) {
        float v = src[(size_t)(r0 + r) * NCH + c];
        acc += v * v;
    }
    atomicAdd(&dst[c], acc);
}

// ysc[c] = 1/(||x_c|| * ||y_c||)  (folded into resident y block)
// xnorm[c] = ||x_c||              (recovers y*rny for warped retrieval)
__global__ void scale_kernel(float* ws) {
    int i = threadIdx.x;   // 256 threads, 1 block
    float sx = ws[i], sy = ws[256 + i];
    ws[512 + i] = rsqrtf(sx) * rsqrtf(sy);
    ws[768 + i] = sqrtf(sx);
}

// ---------------------------------------------------------------------------
// Flash-style pass: one workgroup owns a 64-column j-block.
//   Phase 1: stream all 8192 rows in 32-row tiles (async double-buffered raw
//            copies); fp32 WMMA corr tile; row-max -> global atomic max;
//            online column softmax stats in LDS.
//   Phase 2: recompute rows 0..255 and accumulate warped_color.
__global__ __launch_bounds__(256)
void corr_kernel(const float* __restrict__ X, const float* __restrict__ Y,
                 const float* __restrict__ ysc, const float* __restrict__ xnorm,
                 float* __restrict__ warped, float* __restrict__ conf) {
    __shared__ float ys[JT][CP];          // y * rnx * rny block  (~65 KB)
    __shared__ float xs[2][IT][CP];       // raw x tiles, double-buffered (~65 KB)
    __shared__ float cs[IT][JT + 1];      // corr tile            (~8.1 KB)
    __shared__ float mcol[JT];
    __shared__ float scol[JT];
    __shared__ float part[IT][8];

    const int tid = threadIdx.x;
    const int j0  = blockIdx.x * JT;

    // Load + scale resident y block (coalesced over linear index)
    for (int idx = tid; idx < JT * NCH; idx += 256) {
        int j = idx >> 8, c = idx & (NCH - 1);
        ys[j][c] = Y[(size_t)(j0 + j) * NCH + c] * ysc[c];
    }
    if (tid < JT) { mcol[tid] = -__builtin_inff(); scol[tid] = 0.0f; }

    const int wave = tid >> 5, lane = tid & 31;
    const int isub = wave >> 2, jsub = wave & 3;        // 2 x 4 wave grid
    const int mrow = lane & 15;                          // A/B row within tile
    const int kh   = (lane >> 4) << 1;                   // K sub-offset: 0 or 2
    const int crow = (lane < 16) ? 0 : 8;                // C/D layout row base
    const int ccol = lane & 15;

    // chunk geometry for async copies: 32 rows x 64 x 16B chunks = 2048 chunks
    const unsigned xs0 = (unsigned)(uintptr_t)(&xs[0][0][0]);
    const unsigned xs1 = (unsigned)(uintptr_t)(&xs[1][0][0]);

    // prologue: issue tile 0 into buffer 0
    #pragma unroll
    for (int s = 0; s < 8; ++s) {
        int chunk = tid + (s << 8);
        int row = chunk >> 6;
        int col = (chunk & 63) << 2;
        async_ld_b128(xs0 + (unsigned)((row * CP + col) << 2),
                      (const void*)(X + (size_t)row * NCH + col));
    }

    // -------------------- Phase 1: all rows --------------------
    int buf = 0;
    for (int i0 = 0; i0 < NPIX; i0 += IT) {
        wait_async0();       // my async chunks for tile i0 have landed
        __syncthreads();     // everyone's chunks visible; prev readers done

        if (i0 + IT < NPIX) {  // prefetch next tile into other buffer
            unsigned nb = buf ? xs0 : xs1;
            #pragma unroll
            for (int s = 0; s < 8; ++s) {
                int chunk = tid + (s << 8);
                int row = chunk >> 6;
                int col = (chunk & 63) << 2;
                async_ld_b128(nb + (unsigned)((row * CP + col) << 2),
                              (const void*)(X + (size_t)(i0 + IT + row) * NCH + col));
            }
        }

        v8f acc = {0.f,0.f,0.f,0.f,0.f,0.f,0.f,0.f};
        const float* xrow = &xs[buf][isub * 16 + mrow][kh];
        const float* yrow = &ys[jsub * 16 + mrow][kh];
        #pragma unroll 8
        for (int k = 0; k < NCH; k += 4) {
            v2f a = *(const v2f*)(xrow + k);
            v2f b = *(const v2f*)(yrow + k);
            acc = __builtin_amdgcn_wmma_f32_16x16x4_f32(
                false, a, false, b, (short)0, acc, false, false);
        }
        #pragma unroll
        for (int r = 0; r < 8; ++r)
            cs[isub * 16 + crow + r][jsub * 16 + ccol] = acc[r];
        __syncthreads();

        // row max over this j-block -> global confidence (atomic max)
        if (tid < IT) {
            float rm = -__builtin_inff();
            for (int j = 0; j < JT; ++j) rm = fmaxf(rm, cs[tid][j]);
            atomicMaxFloat(&conf[i0 + tid], rm);
        }
        // online column softmax stats (thread t owns column t)
        if (tid < JT) {
            float m = mcol[tid], s = scol[tid];
            float tm = m;
            for (int r = 0; r < IT; ++r) tm = fmaxf(tm, cs[r][tid]);
            float ss = (s == 0.0f) ? 0.0f : s * __expf((m - tm) * TAU_INV);
            for (int r = 0; r < IT; ++r)
                ss += __expf((cs[r][tid] - tm) * TAU_INV);
            mcol[tid] = tm; scol[tid] = ss;
        }
        __syncthreads();
        buf ^= 1;
    }

    // -------------------- Phase 2: rows 0..255 -> warped_color --------------------
    for (int k0 = 0; k0 < NCH; k0 += IT) {
        for (int idx = tid; idx < IT * NCH; idx += 256) {
            int r = idx >> 8, c = idx & (NCH - 1);
            xs[0][r][c] = X[(size_t)(k0 + r) * NCH + c];
        }
        __syncthreads();

        v8f acc = {0.f,0.f,0.f,0.f,0.f,0.f,0.f,0.f};
        const float* xrow = &xs[0][isub * 16 + mrow][kh];
        const float* yrow = &ys[jsub * 16 + mrow][kh];
        #pragma unroll 8
        for (int k = 0; k < NCH; k += 4) {
            v2f a = *(const v2f*)(xrow + k);
            v2f b = *(const v2f*)(yrow + k);
            acc = __builtin_amdgcn_wmma_f32_16x16x4_f32(
                false, a, false, b, (short)0, acc, false, false);
        }
        #pragma unroll
        for (int r = 0; r < 8; ++r)
            cs[isub * 16 + crow + r][jsub * 16 + ccol] = acc[r];
        __syncthreads();

        // warped[k] += sum_j exp((corr-m_j)/tau)/s_j * (ys'[j][k] * ||x_k||)
        {
            int r  = tid >> 3;   // 0..31 (row k0+r of warped)
            int ch = tid & 7;    // 8-column chunk
            float xn = xnorm[k0 + r];
            float p = 0.0f;
            for (int jj = ch * 8; jj < ch * 8 + 8; ++jj) {
                float e = __expf((cs[r][jj] - mcol[jj]) * TAU_INV);
                p += (e / scol[jj]) * ys[jj][k0 + r];
            }
            part[r][ch] = p * xn;
        }
        __syncthreads();
        if (tid < IT) {
            float w = 0.0f;
            #pragma unroll
            for (int ch = 0; ch < 8; ++ch) w += part[tid][ch];
            atomicAdd(&warped[k0 + tid], w);
        }
        __syncthreads();
    }
}

// ---------------------------------------------------------------------------
extern "C" void kernel_launch(void* const* d_in, const int* in_sizes, int n_in,
                              void* d_out, int out_size, void* d_ws, size_t ws_size,
                              hipStream_t stream) {
    const float* X = (const float*)d_in[0];
    const float* Y = (const float*)d_in[1];
    float* out = (float*)d_out;
    float* ws  = (float*)d_ws;

    float* sumx  = ws;         // [256]
    float* sumy  = ws + 256;   // [256]
    float* yscal = ws + 512;   // [256] = rnx*rny
    float* xnorm = ws + 768;   // [256] = ||x_c||
    float* warped = out;          // [256]
    float* conf   = out + 256;    // [8192]

    init_kernel<<<(256 + NPIX + 255) / 256, 256, 0, stream>>>(out, ws);
    colsumsq_kernel<<<dim3(NPIX / 128, 2), 256, 0, stream>>>(X, Y, sumx, sumy);
    scale_kernel<<<1, 256, 0, stream>>>(ws);
    corr_kernel<<<NPIX / JT, 256, 0, stream>>>(X, Y, yscal, xnorm, warped, conf);
}